// ConvCaps_80238579023895
// MI455X (gfx1250) — compile-verified
//
#include <hip/hip_runtime.h>

// ConvCaps (matrix capsules, EM routing) — fused single kernel for gfx1250.
// One workgroup (8 wave32) per output spatial position b in [0, N*p*p).
// Votes are never materialized to HBM: recomputed per EM pass with
// V_WMMA_F32_16X16X4_F32. A-operands come from an LDS-resident im2col pose
// tile (ds_load_b64); B-operands come from k-chunked, transposed LDS staging
// of the transform matrices (coalesced b128 global reads, ds_load_b64 feeds),
// with global_prefetch of the next chunk overlapping compute.

typedef float v2f __attribute__((ext_vector_type(2)));
typedef float v8f __attribute__((ext_vector_type(8)));

#define KSZ     288      // 3*3*32 input capsules per position
#define OSZ     32       // output capsules
#define KC      32       // k-chunk staged in LDS (KC*512 floats = 64KB)
#define EPSF    1e-9f
#define LOG2PI  1.8378770664093453f

// dynamic-LDS partition (in floats)
#define OFF_POSE  0                         // 4608
#define OFF_ACT   (OFF_POSE + KSZ*16)       // 288
#define OFF_R     (OFF_ACT  + KSZ)          // 9216
#define OFF_MU    (OFF_R    + KSZ*OSZ)      // 512
#define OFF_ISG   (OFF_MU   + OSZ*16)       // 512
#define OFF_AL    (OFF_ISG  + OSZ*16)       // 32
#define OFF_W     (OFF_AL   + OSZ)          // KC*512 = 16384
#define SMEM_FLTS (OFF_W + KC*512)          // 31552 floats = 126208 B

__device__ __forceinline__ void stage_w_chunk(float* __restrict__ w_s,
                                              const float4* __restrict__ wmat4,
                                              int k0, int tid)
{
    // source: w[(k,o,c,n)] row-major, float4 = one (k,o,c) covering n=0..3
    // dest (transposed): w_s[kk*512 + o*16 + n*4 + c]  -> B pair (c0,c0+1) contiguous
    const float4* src = wmat4 + (size_t)k0 * 128;
#pragma unroll
    for (int i = 0; i < (KC * 128) / 256; ++i) {
        int idx = tid + i * 256;            // [0, KC*128)
        float4 v = src[idx];
        int kk = idx >> 7;
        int rem = idx & 127;
        int o = rem >> 2, c = rem & 3;
        float* dst = w_s + kk * 512 + o * 16 + c;
        dst[0]  = v.x;
        dst[4]  = v.y;
        dst[8]  = v.z;
        dst[12] = v.w;
    }
}

__global__ __launch_bounds__(256)
void convcaps_em_kernel(const float* __restrict__ pose,    // (N,14,14,32,16)
                        const float* __restrict__ act,     // (N,14,14,32,1)
                        const float* __restrict__ wmat,    // (288,32,4,4)
                        const float* __restrict__ beta_v,  // (32)
                        const float* __restrict__ beta_a,  // (32)
                        float* __restrict__ out_pose,      // (N*p*p,32,16)
                        float* __restrict__ out_act,       // (N*p*p,32)
                        int H, int Wd, int P)
{
    extern __shared__ float smem[];
    float* pose_s = smem + OFF_POSE;
    float* act_s  = smem + OFF_ACT;
    float* r_s    = smem + OFF_R;
    float* mu_s   = smem + OFF_MU;
    float* isg_s  = smem + OFF_ISG;
    float* aL_s   = smem + OFF_AL;
    float* w_s    = smem + OFF_W;

    const float4* wmat4 = reinterpret_cast<const float4*>(wmat);

    const int b   = blockIdx.x;
    const int img = b / (P * P);
    const int ph  = (b / P) % P;
    const int pw  = b % P;
    const int tid = threadIdx.x;

    // ---- im2col gather of the 3x3 (stride 2) capsule neighborhood ----
    for (int idx = tid; idx < KSZ * 16; idx += 256) {
        int kk = idx >> 4, d = idx & 15;
        int ki = kk / 96, kj = (kk / 32) % 3, cc = kk & 31;
        int h = ph * 2 + ki, w2 = pw * 2 + kj;
        pose_s[idx] = pose[((((size_t)img * H + h) * Wd + w2) * 32 + cc) * 16 + d];
    }
    for (int idx = tid; idx < KSZ; idx += 256) {
        int ki = idx / 96, kj = (idx / 32) % 3, cc = idx & 31;
        int h = ph * 2 + ki, w2 = pw * 2 + kj;
        act_s[idx] = act[(((size_t)img * H + h) * Wd + w2) * 32 + cc];
    }
    for (int idx = tid; idx < KSZ * OSZ; idx += 256)
        r_s[idx] = 1.0f / (float)OSZ;

    // lane-role decode (wave32)
    const int  l     = tid & 31;
    const int  wv    = tid >> 5;
    const int  m     = l & 15;             // A row (valid pose rows 0..3)
    const int  c0    = (l >> 4) << 1;      // contraction half: K={c0,c0+1}
    const int  osub  = (l & 15) >> 2;
    const int  n     = l & 3;
    const bool low   = (l < 16);
    const bool arow  = (m < 4);
    const int  abase = (((m & 3) << 2) + c0);      // safe in-bounds A offset
    const int  o     = (wv << 2) + osub;           // M-step output capsule
    const int  bofM  = o * 16 + n * 4 + c0;        // M-step B offset in w_s
    const int  bofE  = osub * 16 + n * 4 + c0;     // E-step B offset base (+g*64)

    const float bv = beta_v[o];
    const float ba = beta_a[o];

    for (int it = 0; it < 3; ++it) {
        const float inv_temp = (it == 0) ? 5.0e-4f : (it == 1) ? 9.75e-4f : 1.42625e-3f;

        // ================= M-step: wave wv owns o in [4wv, 4wv+4) ==========
        float S0 = 0.f, S1[4] = {0.f,0.f,0.f,0.f}, S2[4] = {0.f,0.f,0.f,0.f};
        for (int k0 = 0; k0 < KSZ; k0 += KC) {
            __syncthreads();                       // w_s reuse / r_s, mu_s ready
            stage_w_chunk(w_s, wmat4, k0, tid);
            __syncthreads();
            if (k0 + KC < KSZ)                     // warm L2/L0 for next chunk
                __builtin_prefetch(wmat + (size_t)(k0 + KC) * 512 + (tid << 6), 0, 1);
            for (int kk = 0; kk < KC; ++kk) {
                const int k = k0 + kk;
                float2 ap = *reinterpret_cast<const float2*>(&pose_s[(k << 4) + abase]);
                float2 bp = *reinterpret_cast<const float2*>(&w_s[kk * 512 + bofM]);
                v2f A; A.x = arow ? ap.x : 0.f; A.y = arow ? ap.y : 0.f;
                v2f B; B.x = bp.x;              B.y = bp.y;
                v8f accv = {0.f,0.f,0.f,0.f,0.f,0.f,0.f,0.f};
                accv = __builtin_amdgcn_wmma_f32_16x16x4_f32(false, A, false, B,
                                                             (short)0, accv, false, false);
                float rw = r_s[k * OSZ + o] * act_s[k];
                S0 += rw;
#pragma unroll
                for (int j = 0; j < 4; ++j) {      // votes v[k,o,d], d=j*4+n
                    float v = accv[j];
                    S1[j] += rw * v;
                    S2[j] += rw * v * v;
                }
            }
        }

        // moments -> mu, sigma^2, cost, activation
        float R = S0 + EPSF;
        float rcpR = 1.0f / R;                     // one precise divide, reused 8x
        float mu[4], sg[4], slg = 0.f;
#pragma unroll
        for (int j = 0; j < 4; ++j) {
            mu[j] = S1[j] * rcpR;
            float num = S2[j] - mu[j] * (2.f * S1[j] - mu[j] * S0); // sum rw*(v-mu)^2
            sg[j] = fmaxf(num * rcpR, 0.f) + EPSF;
            slg += __logf(sg[j]);
        }
        slg += __shfl_xor(slg, 1, 32);             // sum over 16 d's: 4 regs x 4 lanes
        slg += __shfl_xor(slg, 2, 32);
        float costsum = R * (16.f * bv + 0.5f * slg);
        float aout = 1.f / (1.f + __expf(-inv_temp * (ba - costsum)));

        if (it == 2) {
            if (low) {
#pragma unroll
                for (int j = 0; j < 4; ++j)
                    out_pose[(((size_t)b * OSZ + o) << 4) + (j << 2) + n] = mu[j];
                if (n == 0) out_act[(size_t)b * OSZ + o] = aout;
            }
        } else {
            if (low) {
#pragma unroll
                for (int j = 0; j < 4; ++j) {
                    mu_s [(o << 4) + (j << 2) + n] = mu[j];
                    isg_s[(o << 4) + (j << 2) + n] = __builtin_amdgcn_rcpf(sg[j]);
                }
                if (n == 0)
                    aL_s[o] = __logf(aout + EPSF) - 0.5f * (16.f * LOG2PI + slg);
            }
        }

        // ================= E-step: per chunk, wave wv owns KC/8 k's ========
        if (it < 2) {
            for (int k0 = 0; k0 < KSZ; k0 += KC) {
                __syncthreads();                   // mu_s/isg_s/aL_s ready; w_s free
                stage_w_chunk(w_s, wmat4, k0, tid);
                __syncthreads();
                if (k0 + KC < KSZ)
                    __builtin_prefetch(wmat + (size_t)(k0 + KC) * 512 + (tid << 6), 0, 1);
                const int kkb = wv * (KC / 8);
#pragma unroll
                for (int i = 0; i < KC / 8; ++i) {
                    const int kk = kkb + i;
                    const int k  = k0 + kk;
                    float2 ap = *reinterpret_cast<const float2*>(&pose_s[(k << 4) + abase]);
                    v2f A; A.x = arow ? ap.x : 0.f; A.y = arow ? ap.y : 0.f;
                    float xv[8];
#pragma unroll
                    for (int g = 0; g < 8; ++g) {
                        const int og = (g << 2) + osub;
                        float2 bp = *reinterpret_cast<const float2*>(
                                        &w_s[kk * 512 + bofE + (g << 6)]);
                        v2f B; B.x = bp.x; B.y = bp.y;
                        v8f accv = {0.f,0.f,0.f,0.f,0.f,0.f,0.f,0.f};
                        accv = __builtin_amdgcn_wmma_f32_16x16x4_f32(false, A, false, B,
                                                                     (short)0, accv, false, false);
                        float t = 0.f;
#pragma unroll
                        for (int j = 0; j < 4; ++j) {
                            float dd = accv[j] - mu_s[(og << 4) + (j << 2) + n];
                            t += dd * dd * isg_s[(og << 4) + (j << 2) + n];
                        }
                        t += __shfl_xor(t, 1, 32); // sum over the 16 d's
                        t += __shfl_xor(t, 2, 32);
                        xv[g] = aL_s[og] - 0.5f * t;   // log(a)+log_p
                    }
                    // softmax over all 32 o's: 8 regs x 4 lane-groups
                    float mx = xv[0];
#pragma unroll
                    for (int g = 1; g < 8; ++g) mx = fmaxf(mx, xv[g]);
                    mx = fmaxf(mx, __shfl_xor(mx, 4, 32));
                    mx = fmaxf(mx, __shfl_xor(mx, 8, 32));
                    float ss = 0.f;
#pragma unroll
                    for (int g = 0; g < 8; ++g) { xv[g] = __expf(xv[g] - mx); ss += xv[g]; }
                    ss += __shfl_xor(ss, 4, 32);
                    ss += __shfl_xor(ss, 8, 32);
                    float inv = __builtin_amdgcn_rcpf(ss);   // v_rcp_f32
                    if (low && n == 0) {
#pragma unroll
                        for (int g = 0; g < 8; ++g)
                            r_s[k * OSZ + (g << 2) + osub] = xv[g] * inv;
                    }
                }
            }
            // next M-step's first chunk barrier orders r_s for all waves
        }
    }
}

extern "C" void kernel_launch(void* const* d_in, const int* in_sizes, int n_in,
                              void* d_out, int out_size, void* d_ws, size_t ws_size,
                              hipStream_t stream) {
    (void)n_in; (void)out_size; (void)d_ws; (void)ws_size;
    const float* pose = (const float*)d_in[0];
    const float* act  = (const float*)d_in[1];
    const float* w    = (const float*)d_in[2];   // (1,288,32,4,4)
    const float* bv   = (const float*)d_in[3];
    const float* ba   = (const float*)d_in[4];

    const int H = 14, Wd = 14, P = (H - 3) / 2 + 1;      // p = 6
    const int N = in_sizes[0] / (H * Wd * 32 * 16);      // 16
    const int B = N * P * P;                             // 576

    float* out_pose = (float*)d_out;
    float* out_act  = out_pose + (size_t)B * OSZ * 16;

    convcaps_em_kernel<<<B, 256, SMEM_FLTS * sizeof(float), stream>>>(
        pose, act, w, bv, ba, out_pose, out_act, H, Wd, P);
}